// FlattenDense_3092376453299
// MI455X (gfx1250) — compile-verified
//
#include <hip/hip_runtime.h>
#include <stdint.h>

typedef __attribute__((ext_vector_type(2))) float v2f;
typedef __attribute__((ext_vector_type(4))) float v4f;
typedef __attribute__((ext_vector_type(8))) float v8f;

// Problem: B=16, H=64, W=64, C=64, UNITS=128
// x[b,h,j,c] strides: 262144, 4096, 64, 1
// w[u,c,i,h] strides: 262144, 4096, 64, 1
// y[b,i,j,u] strides: 524288, 8192, 128, 1
//
// GEMM per i:  Y[u, (b,j)] = sum_k A[u,k] * Bm[k,(b,j)],  k=(c,h), K=4096
// Workgroup tile: 128 u x 64 n (one b), K in KB=64 blocks (8c x 8h),
// k_local = cl*8 + hl. LDS rows padded to 68 floats (conflict-free b64 reads).

#define SA 68
#define SB 68

#if defined(__AMDGCN__) && \
    __has_builtin(__builtin_amdgcn_global_load_async_to_lds_b128) && \
    __has_builtin(__builtin_amdgcn_global_load_async_to_lds_b32)
#define HAVE_ASYNC_LDS 1
typedef int i32x4_gcc __attribute__((vector_size(16)));
typedef __attribute__((address_space(1))) i32x4_gcc g_i32x4;
typedef __attribute__((address_space(1))) int       g_i32;
typedef __attribute__((address_space(3))) i32x4_gcc l_i32x4;
typedef __attribute__((address_space(3))) int       l_i32;
#else
#define HAVE_ASYNC_LDS 0
#endif

__device__ __forceinline__ void async_wait0() {
#if HAVE_ASYNC_LDS
#if __has_builtin(__builtin_amdgcn_s_wait_asynccnt)
  __builtin_amdgcn_s_wait_asynccnt(0);
#else
  asm volatile("s_wait_asynccnt 0x0" ::: "memory");
#endif
#endif
}

__global__ __launch_bounds__(256) void fused_einsum_relu_wmma(
    const float* __restrict__ x, const float* __restrict__ w,
    float* __restrict__ y) {
  __shared__ float As[128 * SA];
  __shared__ float Bs[64 * SB];

  const int t  = threadIdx.x;
  const int i  = blockIdx.x;      // 0..63  (W index of output / w)
  const int bb = blockIdx.y;      // 0..15  (batch; n0 = bb*64)

  const int lane = t & 31;
  const int lm   = lane & 15;     // row/col within 16
  const int lh   = lane >> 4;     // half-select
  const int wv   = t >> 5;        // wave id 0..7
  const int mg   = wv >> 1;       // 0..3 -> m-tiles {2mg, 2mg+1}
  const int ng   = wv & 1;        // 0..1 -> n-tiles {2ng, 2ng+1}

  v8f acc00 = {}, acc01 = {}, acc10 = {}, acc11 = {};

  const float* wI = w + i * 64;                 // + u*262144 + c*4096 + h
  const float* xB = x + bb * 262144;            // + h*4096 + j*64 + c

  const float* a0p = &As[(mg * 32 + lm) * SA];
  const float* a1p = &As[(mg * 32 + 16 + lm) * SA];
  const float* b0p = &Bs[(ng * 32 + lm) * SB];
  const float* b1p = &Bs[(ng * 32 + 16 + lm) * SB];
  const int koff = 2 * lh;

  for (int kb = 0; kb < 64; ++kb) {
    const int c0 = (kb >> 3) * 8;
    const int h0 = (kb & 7) * 8;

    __syncthreads();   // previous compute done before overwriting LDS

#if HAVE_ASYNC_LDS
    // ---- async stage A tile: 128 u x 64 k, b128 memory->LDS ----
#pragma unroll
    for (int q = 0; q < 8; ++q) {
      const int f  = q * 256 + t;
      const int u  = f >> 4;
      const int cl = (f >> 1) & 7;
      const int h4 = (f & 1) * 4;
      __builtin_amdgcn_global_load_async_to_lds_b128(
          (g_i32x4*)(void*)(wI + u * 262144 + (c0 + cl) * 4096 + h0 + h4),
          (l_i32x4*)(void*)&As[u * SA + cl * 8 + h4], 0, 0);
    }
    // ---- async stage B tile: 64 k x 64 n, b32 scatter into [n][k] ----
#pragma unroll
    for (int q = 0; q < 2; ++q) {
      const int g  = q * 256 + t;
      const int n  = g >> 3;
      const int hl = g & 7;
      const float* p = xB + (h0 + hl) * 4096 + n * 64 + c0;
#pragma unroll
      for (int cl = 0; cl < 8; ++cl) {
        __builtin_amdgcn_global_load_async_to_lds_b32(
            (g_i32*)(void*)(p + cl),
            (l_i32*)(void*)&Bs[n * SB + cl * 8 + hl], 0, 0);
      }
    }
#else
    // ---- stage A tile: 128 u x 64 k (contiguous along h) ----
#pragma unroll
    for (int q = 0; q < 8; ++q) {
      const int f  = q * 256 + t;
      const int u  = f >> 4;
      const int cl = (f >> 1) & 7;
      const int h4 = (f & 1) * 4;
      const v4f v = *(const v4f*)(wI + u * 262144 + (c0 + cl) * 4096 + h0 + h4);
      *(v4f*)(&As[u * SA + cl * 8 + h4]) = v;
    }
    // ---- stage B tile: 64 k x 64 n (contiguous along c) ----
#pragma unroll
    for (int q = 0; q < 2; ++q) {
      const int g  = q * 256 + t;
      const int n  = g >> 3;
      const int hl = g & 7;
      const float* p = xB + (h0 + hl) * 4096 + n * 64 + c0;
      const v4f v0 = *(const v4f*)(p);
      const v4f v1 = *(const v4f*)(p + 4);
      float* bq = &Bs[n * SB + hl];
      bq[0]  = v0.x; bq[8]  = v0.y; bq[16] = v0.z; bq[24] = v0.w;
      bq[32] = v1.x; bq[40] = v1.y; bq[48] = v1.z; bq[56] = v1.w;
    }
#endif

    // prefetch next K-block into near caches (locality 3 -> WGP scope)
    if (kb + 1 < 64) {
      const int c0n = ((kb + 1) >> 3) * 8;
      const int h0n = ((kb + 1) & 7) * 8;
      __builtin_prefetch(wI + (t >> 4) * 262144 + (c0n + ((t >> 1) & 7)) * 4096 +
                             h0n + (t & 1) * 4, 0, 3);
      __builtin_prefetch(xB + (h0n + (t & 7)) * 4096 + (t >> 3) * 64 + c0n, 0, 3);
    }

    async_wait0();     // own async LDS writes committed before publishing
    __syncthreads();

    // ---- compute: 16 k-steps of V_WMMA_F32_16X16X4_F32 ----
#pragma unroll
    for (int k0 = 0; k0 < 64; k0 += 4) {
      const v2f a0 = *(const v2f*)(a0p + k0 + koff);
      const v2f a1 = *(const v2f*)(a1p + k0 + koff);
      const v2f b0 = *(const v2f*)(b0p + k0 + koff);
      const v2f b1 = *(const v2f*)(b1p + k0 + koff);
      acc00 = __builtin_amdgcn_wmma_f32_16x16x4_f32(false, a0, false, b0,
                                                    (short)0, acc00, false, false);
      acc01 = __builtin_amdgcn_wmma_f32_16x16x4_f32(false, a0, false, b1,
                                                    (short)0, acc01, false, false);
      acc10 = __builtin_amdgcn_wmma_f32_16x16x4_f32(false, a1, false, b0,
                                                    (short)0, acc10, false, false);
      acc11 = __builtin_amdgcn_wmma_f32_16x16x4_f32(false, a1, false, b1,
                                                    (short)0, acc11, false, false);
    }
  }

  // ---- ReLU + nontemporal store (keep y out of L2; w+x stay resident) ----
  float* yB = y + bb * 524288 + i * 8192;   // + j*128 + u
  v8f accs[2][2] = {{acc00, acc01}, {acc10, acc11}};
#pragma unroll
  for (int mi = 0; mi < 2; ++mi) {
#pragma unroll
    for (int ni = 0; ni < 2; ++ni) {
      v8f a = accs[mi][ni];
#pragma unroll
      for (int r = 0; r < 8; ++r) a[r] = fmaxf(a[r], 0.0f);
      const int u0 = (mg * 2 + mi) * 16 + lh * 8;       // 8 consecutive u
      const int j  = (ng * 2 + ni) * 16 + lm;
      float* p = yB + j * 128 + u0;
      __builtin_nontemporal_store((v4f){a[0], a[1], a[2], a[3]}, (v4f*)(p));
      __builtin_nontemporal_store((v4f){a[4], a[5], a[6], a[7]}, (v4f*)(p + 4));
    }
  }
}

extern "C" void kernel_launch(void* const* d_in, const int* in_sizes, int n_in,
                              void* d_out, int out_size, void* d_ws, size_t ws_size,
                              hipStream_t stream) {
  const float* x = (const float*)d_in[0];   // (16,64,64,64)
  const float* w = (const float*)d_in[1];   // (128,64,64,64)
  float* y = (float*)d_out;                 // (16,64,64,128)
  dim3 grid(64, 16);                        // i  x  batch (n0 = 64*bb)
  fused_einsum_relu_wmma<<<grid, 256, 0, stream>>>(x, w, y);
}